// ImplicitLongConvolution2D_53652731462257
// MI455X (gfx1250) — compile-verified
//
#include <hip/hip_runtime.h>
#include <cstddef>
#include <cstdint>

// ---------------------------------------------------------------------------
// ImplicitLongConvolution2D on MI455X (gfx1250, wave32, WMMA).
//
// y = irfft2( rfft2(x, 256x256) * rfft2(k, 256x256), norm='forward')[:128,:128] + x
// with k = MLP(coords).  Implemented as an all-LDS 2D FFT per (b,c) image,
// where every 256-point DFT is a radix-16x16 four-step:
//     X[16*k1+k2] = sum_{n1} w16^{n1 k1} * w256^{n1 k2} * sum_{n2} w16^{n2 k2} x[16 n2 + n1]
// i.e. two complex 16x16x16 matmuls -> 32 x v_wmma_f32_16x16x4_f32 per FFT.
// ---------------------------------------------------------------------------

#define DEVINL __device__ __forceinline__

typedef float v2f __attribute__((ext_vector_type(2)));
typedef float v8f __attribute__((ext_vector_type(8)));

static constexpr int   Hc      = 128;    // image H (=W)
static constexpr int   Cc      = 96;     // channels
static constexpr int   Bc      = 16;     // batch
static constexpr int   Nfft    = 256;    // padded FFT size per axis
static constexpr int   LDR     = 258;    // LDS row stride (float2) -> bank-conflict-free column walk
static constexpr int   SCR     = 272;    // per-wave staging tile: 16 rows * stride 17 (float2)
static constexpr int   NW      = 8;      // waves per workgroup
static constexpr int   NTHR    = NW * 32;
static constexpr float PI2     = 6.28318530717958647692f;

// D = A(16x4) * B(4x16) + C  --  fp32 WMMA, the only f32 matrix op on CDNA5.
static DEVINL v8f wmma4(v2f a, v2f b, v8f c) {
  return __builtin_amdgcn_wmma_f32_16x16x4_f32(false, a, false, b, (short)0, c,
                                               false, false);
}

// ---------------------------------------------------------------------------
// One 256-point complex FFT (SIGN=-1 fwd, +1 inv; both unnormalized, matching
// rfft2 default / irfft2 norm='forward'), executed cooperatively by ONE wave32.
//
// Operand lane mappings (ISA 7.12.2, 32-bit, wave32):
//   A (16x4):  lane l, vgpr v: (M = l&15, K = 2*(l>>4)+v)        [chunk t: K += 4t]
//   B (4x16):  lane l, vgpr v: (K = 2*(l>>4)+v, N = l&15)        [chunk t: K += 4t]
//   C/D:       lane l, vgpr j: (M = j + 8*(l>>4), N = l&15)
//
// `sc` is a per-wave LDS staging tile (16 rows, stride 17 float2) used to
// convert the first product from C/D layout into A-operand layout.
// ---------------------------------------------------------------------------
template <int SIGN, class LoadF, class StoreF>
static DEVINL void fft256_wave(LoadF load, StoreF store, float2* __restrict__ sc) {
  const int lane = (int)(threadIdx.x & 31u);
  const int m    = lane & 15;   // N index of B/CD operands, M index of A operand
  const int h    = lane >> 4;   // half-wave select

  // --- F16 DFT factor, valid as both A operand (M=m,K) and B operand (K,N=m):
  //     value = exp(SIGN * i * 2pi * (m*K)/16)
  v2f Fr[4], Fi[4], FiN[4];
#pragma unroll
  for (int t = 0; t < 4; ++t) {
#pragma unroll
    for (int v = 0; v < 2; ++v) {
      const int   K   = 4 * t + 2 * h + v;
      const float ang = (float)SIGN * (PI2 / 16.0f) * (float)((m * K) & 15);
      float s, c;
      __sincosf(ang, &s, &c);
      Fr[t][v] = c; Fi[t][v] = s; FiN[t][v] = -s;
    }
  }

  // --- Step 1: gather data matrix A_data[n2][n1] = x[16*n2 + n1] as B operand.
  v2f Br[4], Bi[4];
#pragma unroll
  for (int t = 0; t < 4; ++t) {
#pragma unroll
    for (int v = 0; v < 2; ++v) {
      const int    n2 = 4 * t + 2 * h + v;
      const float2 d  = load(16 * n2 + m);
      Br[t][v] = d.x; Bi[t][v] = d.y;
    }
  }

  // --- Step 2: G = F16 * A_data  (complex 16x16x16 -> 16 wmma).
  v8f gr = {0, 0, 0, 0, 0, 0, 0, 0};
  v8f gi = {0, 0, 0, 0, 0, 0, 0, 0};
#pragma unroll
  for (int t = 0; t < 4; ++t) {
    gr = wmma4(Fr[t],  Br[t], gr);
    gr = wmma4(FiN[t], Bi[t], gr);
    gi = wmma4(Fr[t],  Bi[t], gi);
    gi = wmma4(Fi[t],  Br[t], gi);
  }

  // --- Step 3: twiddle H[k2][n1] = G * exp(SIGN i 2pi k2 n1 / 256); stage to LDS.
#pragma unroll
  for (int j = 0; j < 8; ++j) {
    const int   k2  = j + 8 * h;
    const float ang = (float)SIGN * (PI2 / 256.0f) * (float)(k2 * m);
    float s, c;
    __sincosf(ang, &s, &c);
    const float hr = gr[j] * c - gi[j] * s;
    const float hi = gr[j] * s + gi[j] * c;
    sc[k2 * 17 + m] = make_float2(hr, hi);   // (k2, n1) with padded stride 17
  }
  __builtin_amdgcn_wave_barrier();
  asm volatile("" ::: "memory");   // same-wave LDS is in-order; just pin the schedule

  // --- Step 4: reload H as A operand (M=k2 -> lane m, K=n1).
  v2f Ar[4], Ai[4], AiN[4];
#pragma unroll
  for (int t = 0; t < 4; ++t) {
#pragma unroll
    for (int v = 0; v < 2; ++v) {
      const int    K = 4 * t + 2 * h + v;
      const float2 d = sc[m * 17 + K];
      Ar[t][v] = d.x; Ai[t][v] = d.y; AiN[t][v] = -d.y;
    }
  }

  // --- Step 5: R = H * F16  (second complex matmul, 16 wmma).
  v8f rr = {0, 0, 0, 0, 0, 0, 0, 0};
  v8f ri = {0, 0, 0, 0, 0, 0, 0, 0};
#pragma unroll
  for (int t = 0; t < 4; ++t) {
    rr = wmma4(Ar[t],  Fr[t], rr);
    rr = wmma4(AiN[t], Fi[t], rr);
    ri = wmma4(Ar[t],  Fi[t], ri);
    ri = wmma4(Ai[t],  Fr[t], ri);
  }

  // --- Step 6: scatter X[16*k1 + k2], with (k2 = j+8h, k1 = m) in C/D layout.
#pragma unroll
  for (int j = 0; j < 8; ++j) {
    store(16 * m + 8 * h + j, make_float2(rr[j], ri[j]));
  }
}

// ---------------------------------------------------------------------------
// Kernel 1: build K_f[c][kc][kr] (full 256x256 complex spectrum per channel).
// The MLP (2 -> 16 relu -> 96) is evaluated inline in the row-FFT loader.
// One workgroup per channel.  LDS: 128x258 float2 spectrum + 8x272 staging.
// ---------------------------------------------------------------------------
__global__ __launch_bounds__(NTHR) void kf_build_kernel(
    const float* __restrict__ W1, const float* __restrict__ b1,
    const float* __restrict__ W2, const float* __restrict__ b2,
    float2* __restrict__ Kf) {
  extern __shared__ float2 smem[];
  float2* data = smem;                                   // [128][258]
  const int wave = (int)(threadIdx.x >> 5);
  float2* sc = smem + Hc * LDR + wave * SCR;             // per-wave staging

  const int c = (int)blockIdx.x;

  // Phase A: 128 forward row FFTs of the implicit kernel (zero-padded to 256).
  for (int r = wave; r < Hc; r += NW) {
    fft256_wave<-1>(
        [&](int n) -> float2 {
          if (n >= Hc) return make_float2(0.f, 0.f);
          const float gy  = (float)r * (1.0f / 127.0f);  // linspace(0,1,128)
          const float gx  = (float)n * (1.0f / 127.0f);
          float       acc = b2[c];
#pragma unroll
          for (int j = 0; j < 16; ++j) {
            float hv = fmaf(gy, W1[j], fmaf(gx, W1[16 + j], b1[j]));
            hv       = fmaxf(hv, 0.f);
            acc      = fmaf(hv, W2[j * Cc + c], acc);
          }
          return make_float2(acc, 0.f);
        },
        [&](int k, float2 v) { data[r * LDR + k] = v; }, sc);
  }
  __syncthreads();

  // Phase B: 256 forward column FFTs (rows 128..255 are zero) -> global K_f.
  const size_t base = (size_t)c * (size_t)(Nfft * Nfft);
  for (int kc = wave; kc < Nfft; kc += NW) {
    fft256_wave<-1>(
        [&](int n) -> float2 {
          return (n < Hc) ? data[n * LDR + kc] : make_float2(0.f, 0.f);
        },
        [&](int kr, float2 v) { Kf[base + (size_t)kc * Nfft + kr] = v; }, sc);
  }
}

// ---------------------------------------------------------------------------
// Kernel 2: per (b,c) image: rows fwd-FFT -> per-column (fwd-FFT ⊙ K_f ⊙
// inv-FFT, fused in registers/staging LDS) -> rows inv-FFT + crop + residual.
// One workgroup per image (1536 total).
// ---------------------------------------------------------------------------
__global__ __launch_bounds__(NTHR) void fftconv_kernel(
    const float* __restrict__ x, const float2* __restrict__ Kf,
    float* __restrict__ out) {
  extern __shared__ float2 smem[];
  float2* data = smem;                                   // [128][258]
  const int wave = (int)(threadIdx.x >> 5);
  float2* sc = smem + Hc * LDR + wave * SCR;

  const int    img  = (int)blockIdx.x;          // b*Cc + c
  const int    c    = img % Cc;
  const float* xim  = x + (size_t)img * (size_t)(Hc * Hc);
  float*       oim  = out + (size_t)img * (size_t)(Hc * Hc);
  const size_t kfb  = (size_t)c * (size_t)(Nfft * Nfft);

  // Phase A: forward row FFTs, loading x straight from global (coalesced in m).
  for (int r = wave; r < Hc; r += NW) {
    fft256_wave<-1>(
        [&](int n) -> float2 {
          return (n < Hc) ? make_float2(xim[r * Hc + n], 0.f)
                          : make_float2(0.f, 0.f);
        },
        [&](int k, float2 v) { data[r * LDR + k] = v; }, sc);
  }
  __syncthreads();

  // Phase B: per column kc -- forward FFT, pointwise * K_f (L2-resident),
  // inverse FFT; only rows 0..127 of the result are needed (final crop).
  for (int kc = wave; kc < Nfft; kc += NW) {
    fft256_wave<-1>(
        [&](int n) -> float2 {
          return (n < Hc) ? data[n * LDR + kc] : make_float2(0.f, 0.f);
        },
        [&](int kr, float2 v) {
          const float2 kf = Kf[kfb + (size_t)kc * Nfft + kr];
          // complex product, staged (padded) for the inverse FFT's B-gather
          sc[(kr >> 4) * 17 + (kr & 15)] =
              make_float2(v.x * kf.x - v.y * kf.y, v.x * kf.y + v.y * kf.x);
        },
        sc);
    asm volatile("" ::: "memory");
    fft256_wave<1>(
        [&](int n) -> float2 { return sc[(n >> 4) * 17 + (n & 15)]; },
        [&](int nr, float2 v) {
          if (nr < Hc) data[nr * LDR + kc] = v;
        },
        sc);
  }
  __syncthreads();

  // Phase C: inverse row FFTs; keep real part of first 128 cols, add residual.
  for (int r = wave; r < Hc; r += NW) {
    fft256_wave<1>(
        [&](int n) -> float2 { return data[r * LDR + n]; },
        [&](int nc, float2 v) {
          if (nc < Hc) {
            const int idx = r * Hc + nc;
            oim[idx] = v.x + xim[idx];
          }
        },
        sc);
  }
}

// ---------------------------------------------------------------------------
// Host launcher.  Inputs: x(16,96,128,128) f32, W1(2,16), b1(16), W2(16,96),
// b2(96).  d_ws holds K_f: 96 * 256 * 256 * sizeof(float2) = 50,331,648 B.
// ---------------------------------------------------------------------------
extern "C" void kernel_launch(void* const* d_in, const int* in_sizes, int n_in,
                              void* d_out, int out_size, void* d_ws,
                              size_t ws_size, hipStream_t stream) {
  (void)in_sizes; (void)n_in; (void)out_size; (void)ws_size;
  const float* x  = (const float*)d_in[0];
  const float* W1 = (const float*)d_in[1];
  const float* b1 = (const float*)d_in[2];
  const float* W2 = (const float*)d_in[3];
  const float* b2 = (const float*)d_in[4];
  float2*      Kf = (float2*)d_ws;
  float*       y  = (float*)d_out;

  const size_t lds_bytes = (size_t)(Hc * LDR + NW * SCR) * sizeof(float2); // 281,600 B

  kf_build_kernel<<<dim3(Cc), dim3(NTHR), lds_bytes, stream>>>(W1, b1, W2, b2, Kf);
  fftconv_kernel<<<dim3(Bc * Cc), dim3(NTHR), lds_bytes, stream>>>(x, Kf, y);
}